// MHCCudaOps_90237262889794
// MI455X (gfx1250) — compile-verified
//
#include <hip/hip_runtime.h>
#include <stdint.h>

// Problem constants (match reference).
#define M_DIM   16384
#define N_DIM   4
#define D_DIM   4096

// Tiling.
#define MT        4                    // m-rows per workgroup
#define DC        512                  // d elements per chunk
#define NCHUNK    (D_DIM / DC)         // 8 chunks
#define THREADS   256                  // 8 wave32 waves
#define LDS_V     1024                 // 16B packets per LDS buffer: 4m*4n*512d*2B / 16B
#define LPT       4                    // async b128 loads per thread per chunk

typedef __bf16 bf16x8 __attribute__((ext_vector_type(8)));   // 16B packet

// Issue one chunk's worth of async global->LDS b128 copies for this thread.
// GVS addressing: SGPR64 base + per-lane 32-bit byte offset (res is 512MiB, fits u32).
// th:TH_LOAD_NT — res is streamed exactly once; don't thrash the 192MB L2.
__device__ __forceinline__ void async_issue_chunk(const uint8_t* __restrict__ res_base,
                                                  uint32_t g_wg_base,
                                                  int c,
                                                  uint32_t lds_buf_byte,
                                                  int tid) {
#pragma unroll
    for (int t = 0; t < LPT; ++t) {
        int l   = t * THREADS + tid;       // 0..1023 : which 16B packet of the tile
        int row = l >> 6;                  // 0..15 == m_local*4 + n  (N_DIM==4)
        int col = l & 63;                  // 16B packet within the 1KB d-row segment
        uint32_t goff = g_wg_base
                      + (uint32_t)row * (uint32_t)(D_DIM * 2)
                      + (uint32_t)c   * (uint32_t)(DC * 2)
                      + (uint32_t)col * 16u;
        uint32_t loff = lds_buf_byte + (uint32_t)row * (uint32_t)(DC * 2) + (uint32_t)col * 16u;
        asm volatile("global_load_async_to_lds_b128 %0, %1, %2 th:TH_LOAD_NT"
                     :: "v"(loff), "v"(goff), "s"(res_base)
                     : "memory");
    }
}

__global__ __launch_bounds__(THREADS)
void mhc_weighted_sum_kernel(const uint8_t* __restrict__ res_bytes,   // [M,N,D] bf16
                             const float4*  __restrict__ h4,          // [M] float4 (N==4)
                             bf16x8*        __restrict__ out8) {      // [M, D/8] bf16 x8
    __shared__ bf16x8 smemv[2 * LDS_V];   // 32 KB, double buffered

    const int tid    = threadIdx.x;
    const int m_base = blockIdx.x * MT;

    const uint32_t g_wg_base = (uint32_t)m_base * (uint32_t)(N_DIM * D_DIM * 2);
    const uint32_t lds0      = (uint32_t)(uintptr_t)&smemv[0];   // LDS byte address

    // Per-thread output coordinates: m_local in [0,4), j indexes an 8-wide d packet.
    const int m_local = tid >> 6;
    const int j       = tid & 63;

    const float4 hv = h4[m_base + m_local];
    const float  hc[4] = {hv.x, hv.y, hv.z, hv.w};

    // Prologue: fill buffer 0.
    async_issue_chunk(res_bytes, g_wg_base, 0, lds0, tid);

    for (int c = 0; c < NCHUNK; ++c) {
        const int buf = c & 1;

        if (c + 1 < NCHUNK) {
            // Keep the next chunk in flight while computing this one.
            async_issue_chunk(res_bytes, g_wg_base, c + 1,
                              lds0 + (uint32_t)((c + 1) & 1) * (uint32_t)(LDS_V * 16), tid);
            // Async loads complete in order: <=4 outstanding => chunk c has landed in LDS.
            asm volatile("s_wait_asynccnt 0x4" ::: "memory");
        } else {
            asm volatile("s_wait_asynccnt 0x0" ::: "memory");
        }
        __syncthreads();   // all waves' async LDS writes visible

        float acc[8];
#pragma unroll
        for (int i = 0; i < 8; ++i) acc[i] = 0.0f;

#pragma unroll
        for (int n = 0; n < N_DIM; ++n) {
            // ds_load_b128: 8 bf16 of res[m_base+m_local][n][c*512 + j*8 ..]
            const bf16x8 q = smemv[buf * LDS_V + (m_local * 4 + n) * 64 + j];
            const float  w = hc[n];
#pragma unroll
            for (int i = 0; i < 8; ++i) {
                // fpext(bf16) feeding fma -> V_FMA_MIX_F32_BF16 (1 VALU per MAC)
                acc[i] = fmaf(w, (float)q[i], acc[i]);
            }
        }

        // f32 -> bf16 RNE (fptrunc) -> v_cvt_pk_bf16_f32 pairs.
        bf16x8 r;
#pragma unroll
        for (int i = 0; i < 8; ++i) r[i] = (__bf16)acc[i];

        // Coalesced nontemporal b128 store: out written once, never re-read.
        __builtin_nontemporal_store(r, &out8[(m_base + m_local) * (D_DIM / 8) + c * (DC / 8) + j]);

        __syncthreads();   // WAR: everyone done reading buf before it is refilled
    }
}

extern "C" void kernel_launch(void* const* d_in, const int* in_sizes, int n_in,
                              void* d_out, int out_size, void* d_ws, size_t ws_size,
                              hipStream_t stream) {
    (void)in_sizes; (void)n_in; (void)out_size; (void)d_ws; (void)ws_size;
    const uint8_t* res = (const uint8_t*)d_in[0];   // bf16 [M,N,D]
    const float4*  h4  = (const float4*)d_in[1];    // f32  [M,N], N==4 -> float4 rows
    bf16x8*        out = (bf16x8*)d_out;            // bf16 [M,D] as 16B packets

    dim3 grid(M_DIM / MT);   // 4096 workgroups
    dim3 block(THREADS);     // 256 threads = 8 waves
    mhc_weighted_sum_kernel<<<grid, block, 0, stream>>>(res, h4, out);
}